// FusedGATOp_20968030339291
// MI455X (gfx1250) — compile-verified
//
#include <hip/hip_runtime.h>
#include <cstdint>
#include <cstddef>

typedef __attribute__((ext_vector_type(2))) float v2f;
typedef __attribute__((ext_vector_type(8))) float v8f;

#define HEADS 4
#define FDIM 32
#define DEG 16
#define HF (HEADS * FDIM)        // 128 floats per node row
#define WAVES_PER_BLOCK 4
#define ROW_PAD 144              // 128 + 16 floats: halves land 32 banks apart -> conflict free

__global__ __launch_bounds__(32 * WAVES_PER_BLOCK)
void gat_fused_kernel(const float* __restrict__ attn_row,
                      const float* __restrict__ attn_col,
                      const int*   __restrict__ rowptr,
                      const int*   __restrict__ colind,
                      const float* __restrict__ neg_slope_p,
                      const float* __restrict__ in_feat,
                      float*       __restrict__ out,
                      int n)
{
    __shared__ __attribute__((aligned(16))) float feat_s[WAVES_PER_BLOCK][DEG][ROW_PAD];

    const int tid    = threadIdx.x;
    const int wid    = tid >> 5;
    const int lane   = tid & 31;
    const int lanelo = lane & 15;
    const int half   = lane >> 4;
    const int node   = blockIdx.x * WAVES_PER_BLOCK + wid;
    if (node >= n) return;                 // wave-uniform: EXEC stays all-ones below

    const float ns = neg_slope_p[0];

    // ---------------- edge scores: lanes 0..15 each own one neighbor ----------------
    const int ebase = rowptr[node];
    int src = 0;
    if (lane < DEG) src = colind[ebase + lane];

    const float4 arow = *(const float4*)(attn_row + (size_t)node * HEADS);
    float4 acol = make_float4(0.f, 0.f, 0.f, 0.f);
    if (lane < DEG) acol = *(const float4*)(attn_col + (size_t)src * HEADS);

    float sc[HEADS];
    sc[0] = arow.x + acol.x;
    sc[1] = arow.y + acol.y;
    sc[2] = arow.z + acol.z;
    sc[3] = arow.w + acol.w;
#pragma unroll
    for (int h = 0; h < HEADS; ++h) {
        float v = sc[h];
        sc[h] = (v > 0.f) ? v : ns * v;
    }

    // ---------------- async gather: 16 neighbor rows (512B each) -> LDS ----------------
    // one wave32 x b128 async op moves one full neighbor row, no VGPR staging
#pragma unroll
    for (int k = 0; k < DEG; ++k) {
        int sk = __shfl(src, k);
        const float* gp = in_feat + (size_t)sk * HF + lane * 4;
        unsigned lp = (unsigned)(size_t)(&feat_s[wid][k][lane * 4]);
        asm volatile("global_load_async_to_lds_b128 %0, %1, off"
                     :: "v"(lp), "v"(gp) : "memory");
    }

    // ---------------- 16-way softmax per head (shfl masks <=8 stay in-half) ----------------
    float mx[HEADS];
#pragma unroll
    for (int h = 0; h < HEADS; ++h) mx[h] = sc[h];
#pragma unroll
    for (int off = 8; off >= 1; off >>= 1) {
#pragma unroll
        for (int h = 0; h < HEADS; ++h)
            mx[h] = fmaxf(mx[h], __shfl_xor(mx[h], off));
    }
    float al[HEADS];
#pragma unroll
    for (int h = 0; h < HEADS; ++h) {
        float m = __builtin_isfinite(mx[h]) ? mx[h] : 0.f;
        al[h] = __expf(sc[h] - m);
    }
    float z[HEADS];
#pragma unroll
    for (int h = 0; h < HEADS; ++h) z[h] = al[h];
#pragma unroll
    for (int off = 8; off >= 1; off >>= 1) {
#pragma unroll
        for (int h = 0; h < HEADS; ++h)
            z[h] += __shfl_xor(z[h], off);
    }
    // v_rcp_f32 instead of the IEEE div_scale/div_fmas/div_fixup chain: z >= 1 exp term,
    // softmax weight tolerance >> 1ulp, and it removes ~60 VALU ops from the critical path
#pragma unroll
    for (int h = 0; h < HEADS; ++h)
        al[h] = al[h] * __builtin_amdgcn_rcpf(z[h]);   // alpha, valid in lanes 0..15

    // wait for the async gather before consuming LDS
    asm volatile("s_wait_asynccnt 0" ::: "memory");

    // ---------------- WMMA aggregation: D = A(alpha,bcast over M) x B(feat) ----------------
    // A 16x4 f32: vgpr0 = K(0|2) per half, vgpr1 = K(1|3)  -> two bpermutes of alpha
    // B 4x16 f32: vgpr0 lane = feat[k0][f=n], vgpr1 = feat[k0+1][f=n]
    // D[m,n] = out_h[c*16+n] replicated over m -> element 0 per lane is the result
    v8f acc[HEADS][2];
#pragma unroll
    for (int h = 0; h < HEADS; ++h)
#pragma unroll
        for (int c = 0; c < 2; ++c)
            acc[h][c] = (v8f){0.f, 0.f, 0.f, 0.f, 0.f, 0.f, 0.f, 0.f};

#pragma unroll
    for (int h = 0; h < HEADS; ++h) {
        float ah = al[h];
#pragma unroll
        for (int kb = 0; kb < DEG; kb += 4) {
            const int k0 = kb + 2 * half;
            float a0 = __shfl(ah, k0);
            float a1 = __shfl(ah, k0 + 1);
            v2f A = {a0, a1};
#pragma unroll
            for (int c = 0; c < 2; ++c) {
                v2f B;
                B.x = feat_s[wid][k0][h * FDIM + c * 16 + lanelo];
                B.y = feat_s[wid][k0 + 1][h * FDIM + c * 16 + lanelo];
                acc[h][c] = __builtin_amdgcn_wmma_f32_16x16x4_f32(
                    false, A, false, B, (short)0, acc[h][c], false, false);
            }
        }
    }

    // ---------------- store: half0 takes c=0, half1 takes c=1 -> contiguous 128B per head ----
    float* op = out + (size_t)node * HF;
#pragma unroll
    for (int h = 0; h < HEADS; ++h) {
        float v = (half == 0) ? acc[h][0][0] : acc[h][1][0];
        op[h * FDIM + lane] = v;   // offset = h*32 + lane  (out_h[lane] in both halves)
    }
}

extern "C" void kernel_launch(void* const* d_in, const int* in_sizes, int n_in,
                              void* d_out, int out_size, void* d_ws, size_t ws_size,
                              hipStream_t stream) {
    const float* attn_row = (const float*)d_in[0];
    const float* attn_col = (const float*)d_in[1];
    const int*   rowptr   = (const int*)d_in[2];
    const int*   colind   = (const int*)d_in[3];
    const float* negslope = (const float*)d_in[4];
    const float* in_feat  = (const float*)d_in[5];
    float*       out      = (float*)d_out;

    const int n = in_sizes[2] - 1;   // rowptr has N+1 entries
    const int blocks = (n + WAVES_PER_BLOCK - 1) / WAVES_PER_BLOCK;
    hipLaunchKernelGGL(gat_fused_kernel, dim3(blocks), dim3(32 * WAVES_PER_BLOCK), 0, stream,
                       attn_row, attn_col, rowptr, colind, negslope, in_feat, out, n);
}